// PlanMapDirectionLoss_14465449853370
// MI455X (gfx1250) — compile-verified
//
#include <hip/hip_runtime.h>
#include <math.h>

typedef float v2f __attribute__((ext_vector_type(2)));
typedef float v8f __attribute__((ext_vector_type(8)));

#define B_SZ   2048
#define T_SZ   6
#define V_SZ   100
#define P_SZ   20
#define NPTS   (V_SZ * P_SZ)   // 2000
#define NGRP   (NPTS / 16)     // 125
#define PI_F     3.14159265358979323846f
#define HALFPI_F 1.57079632679489661923f

// One wave (32 lanes) per batch element. 8 waves per 256-thread block.
__global__ __launch_bounds__(256) void pmdl_main(
    const float* __restrict__ ego,    // [B,6,2]
    const float* __restrict__ lane,   // [B,100,20,2] raw (pre-scale)
    const float* __restrict__ score,  // [B,100,3]
    float* __restrict__ partial)      // [B] per-batch sum of |yd| over t
{
  const int wave    = threadIdx.x >> 5;
  const int lane_id = threadIdx.x & 31;
  const int b       = blockIdx.x * 8 + wave;   // grid=256 -> b in [0,2048)

  const float*    eb  = ego   + (size_t)b * (T_SZ * 2);
  const float*    lb  = lane  + (size_t)b * (NPTS * 2);
  const float*    sb  = score + (size_t)b * (V_SZ * 3);
  const float2*   lb2 = (const float2*)lb;
  const unsigned long long* lq = (const unsigned long long*)lb;

  // ---- per-instance "not lane divider" bitmask, built once (uniform) ----
  // nd[v] == 1  ->  mask the instance's points to 1e6
  unsigned long long m01, m23;
  {
    unsigned mm[4];
#pragma unroll
    for (int c = 0; c < 4; ++c) {
      const int v  = c * 32 + lane_id;
      const int vc = (v < V_SZ) ? v : (V_SZ - 1);     // clamp addr, no divergence
      const float s = sb[vc * 3];
      const bool nd = (v < V_SZ) && (s < 0.5f);
      mm[c] = (unsigned)__ballot(nd);
    }
    m01 = (unsigned long long)mm[0] | ((unsigned long long)mm[1] << 32);
    m23 = (unsigned long long)mm[2] | ((unsigned long long)mm[3] << 32);
  }

  // ---- trajectory: cumsum (every lane redundantly; all registers) ----
  float ex[6], ey[6], cx[6], cy[6];
#pragma unroll
  for (int t = 0; t < 6; ++t) { ex[t] = eb[2*t]; ey[t] = eb[2*t+1]; }
  cx[0] = ex[0]; cy[0] = ey[0];
#pragma unroll
  for (int t = 1; t < 6; ++t) { cx[t] = cx[t-1] + ex[t]; cy[t] = cy[t-1] + ey[t]; }

  // ---- build A fragment: row t = [-2px, -2py, 1, |p|^2]  (16x4 f32) ----
  // Layout (ISA 7.12.2): lanes 0-15 -> K=0,1 in a.x,a.y ; lanes 16-31 -> K=2,3
  const int m    = lane_id & 15;
  const int half = lane_id >> 4;
  float pxm = cx[0], pym = cy[0];
#pragma unroll
  for (int t = 1; t < 6; ++t) { pxm = (m == t) ? cx[t] : pxm; pym = (m == t) ? cy[t] : pym; }
  const float pp = fmaf(pxm, pxm, pym * pym);
  v2f afrag;
  afrag.x = half ? 1.0f : (-2.0f * pxm);
  afrag.y = half ? pp   : (-2.0f * pym);

  // ---- stream 125 groups of 16 points through WMMA, track per-lane argmin ----
  float minv[6]; int mini[6];
#pragma unroll
  for (int t = 0; t < 6; ++t) { minv[t] = 3.4e38f; mini[t] = 0; }

  const int n = lane_id & 15;    // column within group (both halves use same point)
#pragma clang loop unroll(disable)
  for (int g = 0; g < NGRP; ++g) {
    const int j = g * 16 + n;    // flattened point index, < 2000
    const int v = j / 20;
    // prefetch the stream ~1KB ahead (global_prefetch_b8)
    __builtin_prefetch((const char*)lb + (size_t)j * 8 + 1024, 0, 1);
    // single unconditional b64 load; select after the fact (no EXEC games)
    const unsigned long long bits = lq[j];
    const float rx = __uint_as_float((unsigned)bits);
    const float ry = __uint_as_float((unsigned)(bits >> 32));
    const bool  nd = (((v & 64) ? m23 : m01) >> (v & 63)) & 1ULL;
    const float qx = nd ? 1e6f : fmaf(rx, 30.0f, -15.0f);
    const float qy = nd ? 1e6f : fmaf(ry, 60.0f, -30.0f);
    const float qq = fmaf(qx, qx, qy * qy);
    // B fragment: col j = [qx, qy, |q|^2, 1]  (4x16 f32)
    v2f bfrag;
    bfrag.x = half ? qq   : qx;  // K=2 : K=0
    bfrag.y = half ? 1.0f : qy;  // K=3 : K=1
    v8f d = {};
    d = __builtin_amdgcn_wmma_f32_16x16x4_f32(false, afrag, false, bfrag,
                                              (short)0, d, false, false);
    // D: VGPR t, lanes 0-15 = (row t, col n) -> squared distance
#pragma unroll
    for (int t = 0; t < 6; ++t) {
      const float dv = d[t];
      const bool better = dv < minv[t];   // strict '<' keeps first (smallest j)
      minv[t] = better ? dv : minv[t];
      mini[t] = better ? j  : mini[t];
    }
  }

  // ---- reduce argmin across lanes 0..15, broadcast to all lanes ----
#pragma unroll
  for (int t = 0; t < 6; ++t) {
    float v = minv[t]; int idx = mini[t];
#pragma unroll
    for (int off = 8; off >= 1; off >>= 1) {
      const float ov = __shfl_xor(v, off, 16);
      const int   oi = __shfl_xor(idx, off, 16);
      if (ov < v || (ov == v && oi < idx)) { v = ov; idx = oi; }
    }
    mini[t] = __shfl(idx, 0);   // winning flattened point -> instance = /20
  }

  // ---- stage 2: exact reference arithmetic on the winning instance ----
  const float tdx = cx[5] - cx[0], tdy = cy[5] - cy[0];
  const bool  static_mask = sqrtf(tdx * tdx + tdy * tdy) < 1.0f;

  float ty[6];
#pragma unroll
  for (int t = 0; t < 5; ++t) ty[t] = atan2f(cy[t+1] - cy[t], cx[t+1] - cx[t]);
  ty[5] = ty[4];

  float loss_sum = 0.0f;
#pragma unroll
  for (int t = 0; t < 6; ++t) {
    const int vstar = mini[t] / 20;
    const float2* q = lb2 + vstar * 20;
    const bool nd = (((vstar & 64) ? m23 : m01) >> (vstar & 63)) & 1ULL;
    float best = 3.4e38f; int bp = 0;
#pragma clang loop unroll(disable)
    for (int p = 0; p < 20; ++p) {
      const float2 r = q[p];
      const float qx = nd ? 1e6f : fmaf(r.x, 30.0f, -15.0f);
      const float qy = nd ? 1e6f : fmaf(r.y, 60.0f, -30.0f);
      const float dx = qx - cx[t], dy = qy - cy[t];
      const float dd = sqrtf(dx * dx + dy * dy);
      if (dd < best) { best = dd; bp = p; }
    }
    const int bpn = (bp == P_SZ - 1) ? (P_SZ - 2) : (bp + 1);
    const float2 r0 = q[bp], r1 = q[bpn];
    const float mpx = nd ? 1e6f : fmaf(r0.x, 30.0f, -15.0f);
    const float mpy = nd ? 1e6f : fmaf(r0.y, 60.0f, -30.0f);
    const float mnx = nd ? 1e6f : fmaf(r1.x, 30.0f, -15.0f);
    const float mny = nd ? 1e6f : fmaf(r1.y, 60.0f, -30.0f);
    const bool  dist_mask = best > 2.0f;
    const float lane_yaw = atan2f(mny - mpy, mnx - mpx);
    float yd = ty[t] - lane_yaw;
    if (yd >  PI_F)     yd -= PI_F;
    if (yd >  HALFPI_F) yd -= PI_F;
    if (yd < -PI_F)     yd += PI_F;
    if (yd < -HALFPI_F) yd += PI_F;
    if (dist_mask)   yd = 0.0f;
    if (static_mask) yd = 0.0f;
    loss_sum += fabsf(yd);
  }

  if (lane_id == 0) partial[b] = loss_sum;
}

// Deterministic final reduction: fixed strides + fixed LDS tree => bit-stable.
__global__ __launch_bounds__(256) void pmdl_reduce(const float* __restrict__ partial,
                                                   float* __restrict__ out)
{
  __shared__ float sm[256];
  float s = 0.0f;
  for (int i = threadIdx.x; i < B_SZ; i += 256) s += partial[i];
  sm[threadIdx.x] = s;
  __syncthreads();
  for (int off = 128; off > 0; off >>= 1) {
    if ((int)threadIdx.x < off) sm[threadIdx.x] += sm[threadIdx.x + off];
    __syncthreads();
  }
  if (threadIdx.x == 0) out[0] = sm[0] * (1.0f / (float)(B_SZ * T_SZ));
}

extern "C" void kernel_launch(void* const* d_in, const int* in_sizes, int n_in,
                              void* d_out, int out_size, void* d_ws, size_t ws_size,
                              hipStream_t stream) {
  const float* ego   = (const float*)d_in[0];   // [2048,6,2]
  const float* lanes = (const float*)d_in[1];   // [2048,100,20,2]
  const float* score = (const float*)d_in[2];   // [2048,100,3]
  float* out     = (float*)d_out;
  float* partial = (float*)d_ws;                // needs 2048 floats = 8 KB

  pmdl_main<<<dim3(B_SZ / 8), dim3(256), 0, stream>>>(ego, lanes, score, partial);
  pmdl_reduce<<<dim3(1), dim3(256), 0, stream>>>(partial, out);
}